// GCN_2319282339912
// MI455X (gfx1250) — compile-verified
//
#include <hip/hip_runtime.h>

typedef __attribute__((ext_vector_type(2))) float v2f;
typedef __attribute__((ext_vector_type(8))) float v8f;

#define N_NODES 100000
#define N_EDGES 1600000
#define IN_DIM 256
#define HID_DIM 128
#define OUT_DIM 64

// ---------------------------------------------------------------------------
// GEMM1: H1[N,128] = X[N,256] @ W1[256,128], fp32 WMMA 16x16x4
// Block: 256 threads = 8 waves; each wave owns one 16-col N tile of a 16-row
// M tile. X tile staged once in LDS, W1 staged in 64-row K chunks.
// ---------------------------------------------------------------------------
__global__ __launch_bounds__(256) void gemm1_wmma(const float* __restrict__ X,
                                                  const float* __restrict__ W1,
                                                  float* __restrict__ H1) {
    __shared__ float xs[16][260];   // pad 260: banks (4*row + k) distinct
    __shared__ float w1s[64][136];  // pad 136: half-wave k+2 shifts banks by 16

    const int tid  = threadIdx.x;
    const int lane = tid & 31;
    const int wave = tid >> 5;       // 0..7 -> N tile index
    const int half = lane >> 4;      // 0: lanes 0-15, 1: lanes 16-31
    const int l16  = lane & 15;
    const int mbase = blockIdx.x * 16;

    // Stage X tile [16][256]: 16 floats (4x float4) per thread.
    {
        int row = tid >> 4;           // 0..15
        int seg = (tid & 15) * 16;    // 0,16,...,240
        const float* xp = X + (size_t)(mbase + row) * IN_DIM + seg;
        float4 a0 = ((const float4*)xp)[0];
        float4 a1 = ((const float4*)xp)[1];
        float4 a2 = ((const float4*)xp)[2];
        float4 a3 = ((const float4*)xp)[3];
        *(float4*)&xs[row][seg + 0]  = a0;
        *(float4*)&xs[row][seg + 4]  = a1;
        *(float4*)&xs[row][seg + 8]  = a2;
        *(float4*)&xs[row][seg + 12] = a3;
    }

    v8f acc = {};
    const int nb = wave * 16 + l16;

    for (int kc = 0; kc < IN_DIM; kc += 64) {
        // Stage W1 chunk [64][128]: 2048 float4 across 256 threads.
        for (int i = tid; i < (64 * 128) / 4; i += 256) {
            int flat = i * 4;
            int kr   = flat >> 7;     // /128
            int col  = flat & 127;
            float4 w = *(const float4*)&W1[(size_t)(kc + kr) * HID_DIM + col];
            *(float4*)&w1s[kr][col] = w;
        }
        __syncthreads();

#pragma unroll
        for (int kk = 0; kk < 64; kk += 4) {
            const int ka = kc + kk + half * 2;  // column within xs (full K)
            const int kb = kk + half * 2;       // row within w1s chunk
            v2f a = {xs[l16][ka], xs[l16][ka + 1]};
            v2f b = {w1s[kb][nb], w1s[kb + 1][nb]};
            acc = __builtin_amdgcn_wmma_f32_16x16x4_f32(
                false, a, false, b, (short)0, acc, false, false);
        }
        __syncthreads();
    }

    // D layout: VGPR v -> row mbase + v (+8 for upper half-wave), col = nb.
#pragma unroll
    for (int v = 0; v < 8; ++v) {
        int row = mbase + v + half * 8;
        H1[(size_t)row * HID_DIM + nb] = acc[v];
    }
}

// ---------------------------------------------------------------------------
// GEMM2: H2[N,64] = relu(Hagg[N,128]) @ W2[128,64], fp32 WMMA 16x16x4
// Block: 256 threads = 8 waves = 2 M tiles x 4 N tiles (32 rows x 64 cols).
// ReLU fused into the LDS staging of Hagg. Full W2 staged in LDS.
// ---------------------------------------------------------------------------
__global__ __launch_bounds__(256) void gemm2_wmma(const float* __restrict__ Hagg,
                                                  const float* __restrict__ W2,
                                                  float* __restrict__ H2) {
    __shared__ float hs[32][132];
    __shared__ float w2s[128][72];

    const int tid  = threadIdx.x;
    const int lane = tid & 31;
    const int wave = tid >> 5;
    const int mt   = wave >> 2;     // 0..1
    const int nt   = wave & 3;      // 0..3
    const int half = lane >> 4;
    const int l16  = lane & 15;
    const int mbase = blockIdx.x * 32;

    // Stage relu(Hagg) tile [32][128]: 1024 float4.
    for (int i = tid; i < 1024; i += 256) {
        int flat = i * 4;
        int row  = flat >> 7;
        int col  = flat & 127;
        float4 h = *(const float4*)&Hagg[(size_t)(mbase + row) * HID_DIM + col];
        h.x = fmaxf(h.x, 0.0f);
        h.y = fmaxf(h.y, 0.0f);
        h.z = fmaxf(h.z, 0.0f);
        h.w = fmaxf(h.w, 0.0f);
        *(float4*)&hs[row][col] = h;
    }
    // Stage full W2 [128][64]: 2048 float4.
    for (int i = tid; i < 2048; i += 256) {
        int flat = i * 4;
        int kr   = flat >> 6;
        int col  = flat & 63;
        *(float4*)&w2s[kr][col] = *(const float4*)&W2[(size_t)kr * OUT_DIM + col];
    }
    __syncthreads();

    v8f acc = {};
    const int mrow = mt * 16 + l16;
    const int nb   = nt * 16 + l16;

#pragma unroll
    for (int kk = 0; kk < HID_DIM; kk += 4) {
        const int k = kk + half * 2;
        v2f a = {hs[mrow][k], hs[mrow][k + 1]};
        v2f b = {w2s[k][nb], w2s[k + 1][nb]};
        acc = __builtin_amdgcn_wmma_f32_16x16x4_f32(
            false, a, false, b, (short)0, acc, false, false);
    }

#pragma unroll
    for (int v = 0; v < 8; ++v) {
        int row = mbase + mt * 16 + v + half * 8;
        H2[(size_t)row * OUT_DIM + nt * 16 + l16] = acc[v];
    }
}

// ---------------------------------------------------------------------------
// SpMM scatter, D=128: one wave per edge, each lane owns a float4 slice.
// out[dst[e]] += vals[e] * H[src[e]]; destination stays L2-resident.
// ---------------------------------------------------------------------------
__global__ __launch_bounds__(256) void spmm_scatter_d128(
    const int* __restrict__ src, const int* __restrict__ dst,
    const float* __restrict__ vals, const float* __restrict__ H,
    float* __restrict__ out, int E) {
    int e = blockIdx.x * 8 + threadIdx.y;
    if (e >= E) return;
    int lane = threadIdx.x;              // 0..31
    int s = src[e];
    int d = dst[e];
    float v = vals[e];
    float4 m = ((const float4*)(H + (size_t)s * 128))[lane];
    float* op = out + (size_t)d * 128 + lane * 4;
    atomicAdd(op + 0, m.x * v);
    atomicAdd(op + 1, m.y * v);
    atomicAdd(op + 2, m.z * v);
    atomicAdd(op + 3, m.w * v);
}

// SpMM scatter, D=64: one wave per edge, each lane owns a float2 slice.
__global__ __launch_bounds__(256) void spmm_scatter_d64(
    const int* __restrict__ src, const int* __restrict__ dst,
    const float* __restrict__ vals, const float* __restrict__ H,
    float* __restrict__ out, int E) {
    int e = blockIdx.x * 8 + threadIdx.y;
    if (e >= E) return;
    int lane = threadIdx.x;
    int s = src[e];
    int d = dst[e];
    float v = vals[e];
    float2 m = ((const float2*)(H + (size_t)s * 64))[lane];
    float* op = out + (size_t)d * 64 + lane * 2;
    atomicAdd(op + 0, m.x * v);
    atomicAdd(op + 1, m.y * v);
}

extern "C" void kernel_launch(void* const* d_in, const int* in_sizes, int n_in,
                              void* d_out, int out_size, void* d_ws, size_t ws_size,
                              hipStream_t stream) {
    const float* X     = (const float*)d_in[0];
    const int*   esrc  = (const int*)d_in[1];
    const int*   edst  = (const int*)d_in[2];
    const float* evals = (const float*)d_in[3];
    const float* W1    = (const float*)d_in[4];
    const float* W2    = (const float*)d_in[5];
    float*       out   = (float*)d_out;

    char* ws = (char*)d_ws;
    const size_t h1Bytes = (size_t)N_NODES * HID_DIM * sizeof(float);  // 51.2 MB
    float* H1   = (float*)(ws);
    float* Hagg = (float*)(ws + h1Bytes);
    float* H2   = (float*)(ws + 2 * h1Bytes);

    // Layer 1
    hipMemsetAsync(Hagg, 0, h1Bytes, stream);
    gemm1_wmma<<<N_NODES / 16, 256, 0, stream>>>(X, W1, H1);
    dim3 eb(32, 8);
    spmm_scatter_d128<<<N_EDGES / 8, eb, 0, stream>>>(esrc, edst, evals, H1, Hagg, N_EDGES);

    // Layer 2 (ReLU fused into gemm2 load)
    hipMemsetAsync(out, 0, (size_t)N_NODES * OUT_DIM * sizeof(float), stream);
    gemm2_wmma<<<N_NODES / 32, 256, 0, stream>>>(Hagg, W2, H2);
    spmm_scatter_d64<<<N_EDGES / 8, eb, 0, stream>>>(esrc, edst, evals, H2, out, N_EDGES);
}